// LSTMModel_org_49692771615280
// MI455X (gfx1250) — compile-verified
//
#include <hip/hip_runtime.h>

// ---------------------------------------------------------------------------
// Fused LSTM (B=8192, T=512, I=5, H=5) + two 5->1 heads for MI455X (gfx1250).
// One wave32 owns 32 batch rows; per timestep the [32x10]@[10x20] gate GEMM is
// done with 12 x V_WMMA_F32_16X16X4_F32 (M-tiles x N-tiles x K-chunks).
// x is staged Global->LDS with double-buffered ASYNC copies (ASYNCcnt) so the
// HBM refill latency is hidden behind the 32-step compute of the prior chunk.
// Async copies share one base address pair and use compile-time immediate
// offsets (ISA applies INST_OFFSET to both the global and LDS sides).
// ---------------------------------------------------------------------------

typedef __attribute__((ext_vector_type(2))) float v2f;
typedef __attribute__((ext_vector_type(4))) float v4f;
typedef __attribute__((ext_vector_type(8))) float v8f;
typedef __attribute__((ext_vector_type(4))) int   v4i;

#define T_SEQ    512
#define IN_SZ    5
#define HID      5
#define NGATE    20          // 4*HID, PyTorch order: i, f, g, o
#define TCHUNK   32          // timesteps staged in LDS per refill
#define XSTR     172         // floats per staged x row (160 used + pad); 688B, 16B aligned
#define XBUF     (32 * XSTR) // one x buffer (floats)
#define GSTR     36          // gate LDS column stride (floats); 144B, 16B aligned,
                             // b128 stores land on disjoint 4-bank spans (conflict-free)

#if __has_builtin(__builtin_amdgcn_global_load_async_to_lds_b128)
#define HAVE_ASYNC_LDS 1
// Builtin signature (per hipcc diagnostic): (v4i as1*, v4i as3*, imm int, imm int)
typedef __attribute__((address_space(1))) v4i global_v4i;
typedef __attribute__((address_space(3))) v4i shared_v4i;
#endif

// Unrolled row copy: N x b128, one shared base address pair, constant offsets.
template <int I>
struct XCopy {
  static __device__ __forceinline__ void run(const float* g, float* l) {
    XCopy<I - 1>::run(g, l);
#ifdef HAVE_ASYNC_LDS
    __builtin_amdgcn_global_load_async_to_lds_b128(
        (global_v4i*)g, (shared_v4i*)l, (I - 1) * 16, 0);
#else
    *(v4f*)(l + (I - 1) * 4) = *(const v4f*)(g + (I - 1) * 4);
#endif
  }
};
template <>
struct XCopy<0> {
  static __device__ __forceinline__ void run(const float*, float*) {}
};

static __device__ __forceinline__ void wait_async0() {
#ifdef HAVE_ASYNC_LDS
#if __has_builtin(__builtin_amdgcn_s_wait_asynccnt)
  __builtin_amdgcn_s_wait_asynccnt(0);
#else
  asm volatile("s_wait_asynccnt 0x0" ::: "memory");
#endif
#endif
}

static __device__ __forceinline__ float fast_tanh(float x) {
#if __has_builtin(__builtin_amdgcn_tanhf)
  return __builtin_amdgcn_tanhf(x);          // v_tanh_f32 on gfx1250
#else
  x = fminf(fmaxf(x, -15.f), 15.f);
  float e = __expf(-2.f * x);
  return (1.f - e) * __builtin_amdgcn_rcpf(1.f + e);
#endif
}
static __device__ __forceinline__ float fast_sigmoid(float x) {
  // sigmoid(x) = 0.5*tanh(x/2) + 0.5
  return fmaf(0.5f, fast_tanh(0.5f * x), 0.5f);
}

__global__ __launch_bounds__(32) void lstm_fused_wmma(
    const float* __restrict__ x,      // [B, T, 5]
    const float* __restrict__ W_ih,   // [20, 5]
    const float* __restrict__ W_hh,   // [20, 5]
    const float* __restrict__ b_ih,   // [20]
    const float* __restrict__ b_hh,   // [20]
    const float* __restrict__ fc1_w,  // [1, 5]
    const float* __restrict__ fc1_b,  // [1]
    const float* __restrict__ fc2_w,  // [1, 5]
    const float* __restrict__ fc2_b,  // [1]
    float* __restrict__ out,          // [2*B] : price | volume
    int Btot)
{
  __shared__ __align__(16) float xs[2 * XBUF];      // double-buffered x chunks
  __shared__ __align__(16) float gs[NGATE * GSTR];  // gate tile, column-major
  __shared__ __align__(16) float hsm[32 * HID];     // h, row-major (stride 5: conflict-free)

  const int l     = threadIdx.x;     // 0..31
  const int m16   = l & 15;          // lane position within 16
  const int half  = l >> 4;          // lane half (K split for A/B operands)
  const int rbase = blockIdx.x * 32; // first batch row of this wave

  // ---- Build B operands once: Wc[k][n], k=0..4 -> W_ih^T, k=5..9 -> W_hh^T, k>=10 -> 0.
  // f32 B-layout (K x 16 per N-tile): lane half splits K in half; VGPR v holds
  // row k = 4*c + 2*half + v (mirrors the A 16x4 table: lanes16-31 hold K2,K3).
  v2f Bop[3][2];
  float bias[2];
  for (int nt = 0; nt < 2; ++nt) {
    const int col = nt * 16 + m16;
    bias[nt] = (col < NGATE) ? (b_ih[col] + b_hh[col]) : 0.f;
    for (int c = 0; c < 3; ++c) {
      float bv0 = 0.f, bv1 = 0.f;
      if (col < NGATE) {
        const int k0 = 4 * c + 2 * half;
        const int k1 = k0 + 1;
        bv0 = (k0 < IN_SZ) ? W_ih[col * IN_SZ + k0]
            : (k0 < IN_SZ + HID) ? W_hh[col * HID + (k0 - IN_SZ)] : 0.f;
        bv1 = (k1 < IN_SZ) ? W_ih[col * IN_SZ + k1]
            : (k1 < IN_SZ + HID) ? W_hh[col * HID + (k1 - IN_SZ)] : 0.f;
      }
      v2f b; b[0] = bv0; b[1] = bv1;
      Bop[c][nt] = b;
    }
  }

  // ---- init state
  float creg[HID], hlast[HID];
  #pragma unroll
  for (int j = 0; j < HID; ++j) { creg[j] = 0.f; hlast[j] = 0.f; hsm[l * HID + j] = 0.f; }

  const float* xrow = x + (size_t)(rbase + l) * T_SEQ * IN_SZ;  // this lane stages row l

  // ---- prime buffer 0 with chunk 0 (each lane copies its row: 40 x b128)
  XCopy<(TCHUNK * IN_SZ) / 4>::run(xrow, &xs[l * XSTR]);
  wait_async0();
  __syncthreads();

  int cur = 0;
  for (int t0 = 0; t0 < T_SEQ; t0 += TCHUNK) {
    // ---- kick off async copy of the NEXT chunk into the other buffer
    if (t0 + TCHUNK < T_SEQ) {
      XCopy<(TCHUNK * IN_SZ) / 4>::run(xrow + (t0 + TCHUNK) * IN_SZ,   // 16B aligned
                                       &xs[(cur ^ 1) * XBUF + l * XSTR]);
    }

    const float* xbuf = &xs[cur * XBUF];
    for (int tl = 0; tl < TCHUNK; ++tl) {
      // ---- assemble A operands (16x4 f32 layout: lane m16 = row, half picks K pair)
      // K columns: 0..4 = x_t, 5..9 = h, 10..11 = 0
      v2f A0[2], A1[2], A2[2];
      #pragma unroll
      for (int mt = 0; mt < 2; ++mt) {
        const int am = mt * 16 + m16;
        const float* xr = &xbuf[am * XSTR + tl * IN_SZ];
        const float* hr = &hsm[am * HID];
        v2f a0, a1, a2;
        if (half == 0) { a0[0]=xr[0]; a0[1]=xr[1]; a1[0]=xr[4]; a1[1]=hr[0]; a2[0]=hr[3]; a2[1]=hr[4]; }
        else           { a0[0]=xr[2]; a0[1]=xr[3]; a1[0]=hr[1]; a1[1]=hr[2]; a2[0]=0.f;   a2[1]=0.f;   }
        A0[mt] = a0; A1[mt] = a1; A2[mt] = a2;
      }

      // ---- 12 WMMAs: gates(+bias) for 32 rows x 20 cols, dump to LDS column-major
      #pragma unroll
      for (int mt = 0; mt < 2; ++mt) {
        #pragma unroll
        for (int nt = 0; nt < 2; ++nt) {
          const float b = bias[nt];
          v8f acc = {b, b, b, b, b, b, b, b};
          acc = __builtin_amdgcn_wmma_f32_16x16x4_f32(false, A0[mt], false, Bop[0][nt],
                                                      (short)0, acc, false, false);
          acc = __builtin_amdgcn_wmma_f32_16x16x4_f32(false, A1[mt], false, Bop[1][nt],
                                                      (short)0, acc, false, false);
          acc = __builtin_amdgcn_wmma_f32_16x16x4_f32(false, A2[mt], false, Bop[2][nt],
                                                      (short)0, acc, false, false);
          const int col = nt * 16 + m16;
          if (col < NGATE) {   // C-layout: lane=col; VGPR v = row (l/16)*8+v within M-tile
            float* gp = &gs[col * GSTR + mt * 16 + half * 8];
            v4f lo; lo[0]=acc[0]; lo[1]=acc[1]; lo[2]=acc[2]; lo[3]=acc[3];
            v4f hi; hi[0]=acc[4]; hi[1]=acc[5]; hi[2]=acc[6]; hi[3]=acc[7];
            *(v4f*)(gp)     = lo;
            *(v4f*)(gp + 4) = hi;
          }
        }
      }
      __syncthreads();   // sync1: gate dump -> gate read

      // ---- elementwise LSTM cell: lane l owns batch row l (conflict-free b32 reads)
      #pragma unroll
      for (int j = 0; j < HID; ++j) {
        const float iv = fast_sigmoid(gs[(0 * HID + j) * GSTR + l]);
        const float fv = fast_sigmoid(gs[(1 * HID + j) * GSTR + l]);
        const float gv = fast_tanh   (gs[(2 * HID + j) * GSTR + l]);
        const float ov = fast_sigmoid(gs[(3 * HID + j) * GSTR + l]);
        const float c  = fmaf(fv, creg[j], iv * gv);
        creg[j]  = c;
        hlast[j] = ov * fast_tanh(c);
      }
      // h feedback (safe without extra barrier: this step's h reads were pre-sync1)
      #pragma unroll
      for (int j = 0; j < HID; ++j) hsm[l * HID + j] = hlast[j];
      __syncthreads();   // sync2: (gate reads + h stores) -> next step's (dump + h reads)
    }

    wait_async0();       // next chunk's async Global->LDS copies complete
    __syncthreads();
    cur ^= 1;
  }

  // ---- heads: price = h.fc1_w + fc1_b ; volume = h.fc2_w + fc2_b
  float p = fc1_b[0], v = fc2_b[0];
  #pragma unroll
  for (int j = 0; j < HID; ++j) { p = fmaf(hlast[j], fc1_w[j], p); v = fmaf(hlast[j], fc2_w[j], v); }
  const int row = rbase + l;
  out[row]        = p;
  out[Btot + row] = v;
}

extern "C" void kernel_launch(void* const* d_in, const int* in_sizes, int n_in,
                              void* d_out, int out_size, void* d_ws, size_t ws_size,
                              hipStream_t stream) {
  (void)n_in; (void)d_ws; (void)ws_size; (void)out_size;
  const float* x     = (const float*)d_in[0];
  const float* W_ih  = (const float*)d_in[1];
  const float* W_hh  = (const float*)d_in[2];
  const float* b_ih  = (const float*)d_in[3];
  const float* b_hh  = (const float*)d_in[4];
  const float* fc1_w = (const float*)d_in[5];
  const float* fc1_b = (const float*)d_in[6];
  const float* fc2_w = (const float*)d_in[7];
  const float* fc2_b = (const float*)d_in[8];
  float* out = (float*)d_out;

  const int B = in_sizes[0] / (T_SEQ * IN_SZ);   // 8192
  const int blocks = B / 32;                     // 256 single-wave workgroups
  lstm_fused_wmma<<<blocks, 32, 0, stream>>>(x, W_ih, W_hh, b_ih, b_hh,
                                             fc1_w, fc1_b, fc2_w, fc2_b, out, B);
}